// TrainablePCEN_47545287967363
// MI455X (gfx1250) — compile-verified
//
#include <hip/hip_runtime.h>
#include <cstdint>
#include <cstddef>

namespace {

constexpr int T_LEN   = 4000;              // time steps per row
constexpr int NTHR    = 256;               // 8 wave32s
constexpr int CHUNK   = 16;                // elements per thread
constexpr int VCHUNKS = T_LEN / CHUNK;     // 250 valid chunks
constexpr int NGROUPS = T_LEN / 4;         // 1000 float4 groups per row
constexpr int NWAVES  = NTHR / 32;
// LDS layout: every 16 words followed by 4 pad words -> 16B-aligned async
// b128 destinations AND per-thread chunk stride of 20 words (80B, 16B aligned).
constexpr int LDSW    = 4096 + 4 * 256;    // 5120 words = 20 KB

constexpr double Sd  = 512.0 / (22050.0 * 0.06);   // ~0.386999
constexpr double Ad  = 1.0 - Sd;
constexpr double A2d = Ad * Ad;
constexpr double A4d = A2d * A2d;
constexpr double A8d = A4d * A4d;
constexpr double A16d = A8d * A8d;

constexpr float SF   = (float)Sd;
constexpr float AF   = (float)Ad;
constexpr float A16F = (float)A16d;
constexpr float EPSF = 1e-6f;

typedef float v4f __attribute__((ext_vector_type(4)));

// Raw hardware transcendentals (v_log_f32 = log2, v_exp_f32 = 2^x).
// All bases in this kernel are strictly positive and well inside the
// normal range, so the guarded libm sequences are unnecessary.
__device__ __forceinline__ float fast_log2(float x) {
#if __has_builtin(__builtin_amdgcn_logf)
  return __builtin_amdgcn_logf(x);
#else
  return log2f(x);
#endif
}
__device__ __forceinline__ float fast_exp2(float x) {
#if __has_builtin(__builtin_amdgcn_exp2f)
  return __builtin_amdgcn_exp2f(x);
#else
  return exp2f(x);
#endif
}

__global__ __launch_bounds__(NTHR) void pcen_scan_kernel(
    const float* __restrict__ mel,
    const float* __restrict__ alpha_p,
    const float* __restrict__ delta_p,
    const float* __restrict__ r_p,
    float* __restrict__ out)
{
  __shared__ float lds[LDSW];
  __shared__ float wAs[NWAVES];
  __shared__ float wCs[NWAVES];

  const int tid  = threadIdx.x;
  const int lane = tid & 31;
  const int wid  = tid >> 5;
  const int row  = blockIdx.x;

  const float* rowp = mel + (size_t)row * T_LEN;

  // ---- Stage the whole row into LDS with async global->LDS (gfx1250) ----
  const uint32_t lbase = (uint32_t)(uintptr_t)(&lds[0]);
#pragma unroll
  for (int it = 0; it < 4; ++it) {
    int g = tid + it * NTHR;
    g = (g < NGROUPS) ? g : (NGROUPS - 1);          // clamp: duplicate copy of
                                                    // identical data, benign
    uint32_t voff  = (uint32_t)g * 16u;             // byte offset within row
    uint32_t laddr = lbase + 16u * ((uint32_t)g + ((uint32_t)g >> 2));
    asm volatile("global_load_async_to_lds_b128 %0, %1, %2"
                 :: "v"(laddr), "v"(voff), "s"(rowp)
                 : "memory");
  }
  asm volatile("s_wait_asynccnt 0x0" ::: "memory");
  __syncthreads();

  // ---- Each thread pulls a contiguous 16-element chunk (stride 20 words) ----
  float x[CHUNK];
  const int base = tid * CHUNK;
#pragma unroll
  for (int j = 0; j < CHUNK; ++j) {
    float v = lds[20 * tid + j];
    x[j] = (base + j < T_LEN) ? v : 0.0f;           // tail chunks: zeros
  }

  // ---- Chunk summary: affine map m -> A*m + C over 16 steps ----
  float Ai, Ci;
  {
    float m = (tid == 0) ? x[0] : SF * x[0];        // M[0] = x[0] exactly
#pragma unroll
    for (int j = 1; j < CHUNK; ++j) m = fmaf(AF, m, SF * x[j]);
    Ci = m;
    Ai = (tid == 0) ? 0.0f : A16F;                  // chunk 0 absorbs the init
  }

  // ---- Inclusive wave32 scan of affine maps: (prev)o(cur) ----
  float A = Ai, C = Ci;
#pragma unroll
  for (int d = 1; d < 32; d <<= 1) {
    float Ap = __shfl_up(A, d, 32);
    float Cp = __shfl_up(C, d, 32);
    if (lane >= d) {
      C = fmaf(A, Cp, C);                           // C' = A*Cprev + C
      A = A * Ap;                                   // A' = Aprev*A
    }
  }

  if (lane == 31) { wAs[wid] = A; wCs[wid] = C; }
  __syncthreads();

  // ---- Exclusive prefix across the 8 waves (tiny serial combine) ----
  float pA = 1.0f, pC = 0.0f;
  for (int w = 0; w < wid; ++w) {
    float qA = wAs[w], qC = wCs[w];
    pC = fmaf(qA, pC, qC);
    pA = pA * qA;
  }
  // Exclusive-within-wave from the inclusive scan
  float eA = __shfl_up(A, 1, 32);
  float eC = __shfl_up(C, 1, 32);
  if (lane == 0) { eA = 1.0f; eC = 0.0f; }
  // Total exclusive prefix applied to m0 = 0 -> incoming state
  const float m_in = fmaf(eA, pC, eC);

  // ---- PCEN parameters (uniform scalar loads) ----
  const float al = fminf(fmaxf(alpha_p[0], 0.01f), 0.99f);
  const float de = fabsf(delta_p[0]) + EPSF;
  const float rr = fminf(fmaxf(r_p[0], 0.01f), 1.0f);
  const float deltaR = fast_exp2(rr * fast_log2(de));

  // ---- Replay recurrence with true carry; compute + store coalesced ----
  float* orow = out + (size_t)row * T_LEN;
  const bool wr = (tid < VCHUNKS);
  float m = m_in;
  float ov[4];
#pragma unroll
  for (int j = 0; j < CHUNK; ++j) {
    if (j == 0) {
      m = (tid == 0) ? x[0] : fmaf(AF, m, SF * x[0]);
    } else {
      m = fmaf(AF, m, SF * x[j]);
    }
    // (eps+M)^(-alpha); base > 0 always -> raw v_log_f32 / v_exp_f32
    const float inv_smooth = fast_exp2(-al * fast_log2(EPSF + m));
    const float o = fast_exp2(rr * fast_log2(fmaf(x[j], inv_smooth, de))) - deltaR;
    ov[j & 3] = o;
    if ((j & 3) == 3 && wr) {
      v4f v4 = {ov[0], ov[1], ov[2], ov[3]};
      // streaming output, never re-read: non-temporal b128 store
      __builtin_nontemporal_store(v4, (v4f*)(orow + base + (j - 3)));
    }
  }
}

} // anonymous namespace

extern "C" void kernel_launch(void* const* d_in, const int* in_sizes, int n_in,
                              void* d_out, int out_size, void* d_ws, size_t ws_size,
                              hipStream_t stream) {
  (void)n_in; (void)out_size; (void)d_ws; (void)ws_size;
  const float* mel   = (const float*)d_in[0];
  const float* alpha = (const float*)d_in[1];
  const float* delta = (const float*)d_in[2];
  const float* r     = (const float*)d_in[3];
  float* outp        = (float*)d_out;

  const int rows = in_sizes[0] / T_LEN;   // 128*128 = 16384 independent series
  pcen_scan_kernel<<<rows, NTHR, 0, stream>>>(mel, alpha, delta, r, outp);
}